// GATBranch_21809843929274
// MI455X (gfx1250) — compile-verified
//
#include <hip/hip_runtime.h>
#include <hip/hip_bf16.h>
#include <math.h>

// ---------------------------------------------------------------------------
// GAT branch for MI455X (gfx1250, wave32).
// Dominant cost: per-edge gather/scatter (~3 GB HBM traffic) -> memory bound.
// Dense N x 256 x 256 transforms use V_WMMA_F32_16X16X4_F32 (full fp32).
// Segment softmax done with 3 atomic passes (uint-key float max, exp-sum, msg).
// ---------------------------------------------------------------------------

typedef float v2f __attribute__((ext_vector_type(2)));
typedef float v8f __attribute__((ext_vector_type(8)));

#define HC 256      // H*C
#define HEADS 4
#define CDIM 64
#define NGRAPH 64

// ---------------- monotonic float<->uint key for atomicMax on floats --------
__device__ __forceinline__ unsigned f2key(float f) {
  unsigned u = __float_as_uint(f);
  return (u & 0x80000000u) ? ~u : (u | 0x80000000u);
}
__device__ __forceinline__ float key2f(unsigned k) {
  unsigned u = (k & 0x80000000u) ? (k & 0x7FFFFFFFu) : ~k;
  return __uint_as_float(u);
}

// ---------------- generic 32-bit fill --------------------------------------
__global__ void fill_u32(unsigned* __restrict__ p, unsigned val, long n) {
  long i = (long)blockIdx.x * blockDim.x + threadIdx.x;
  if (i < n) p[i] = val;
}

// ---------------- layer 1 transform: x[N,2] @ W1[2,256] --------------------
__global__ void layer1_transform(const float* __restrict__ x,
                                 const float* __restrict__ W,
                                 float* __restrict__ h, int Nn) {
  long i = (long)blockIdx.x * blockDim.x + threadIdx.x;   // over N*256
  if (i >= (long)Nn * HC) return;
  int n = (int)(i >> 8);
  int j = (int)(i & (HC - 1));
  h[i] = x[(long)n * 2 + 0] * W[j] + x[(long)n * 2 + 1] * W[HC + j];
}

// ---------------- fp32 WMMA GEMM: C[M,Ncol] = A[M,K] @ B[K,Ncol] -----------
// One wave -> one 16x16 tile, accumulating V_WMMA_F32_16X16X4_F32 over K.
__global__ void gemm_f32_wmma(const float* __restrict__ A,
                              const float* __restrict__ B,
                              float* __restrict__ C,
                              int M, int K, int Ncol) {
  const int wave = threadIdx.x >> 5;       // 8 waves per 256-thread block
  const int lane = threadIdx.x & 31;
  const int tiles_n = Ncol >> 4;
  int tile = blockIdx.x * 8 + wave;
  int tm = tile / tiles_n;
  int tn = tile - tm * tiles_n;
  if (tm * 16 >= M) return;                // wave-uniform exit

  const int row   = tm * 16 + (lane & 15); // A row held by this lane
  const int rcl   = row < M ? row : M - 1; // clamp (N is a multiple of 16 here)
  const int col   = tn * 16 + (lane & 15); // B/C column held by this lane
  const int khalf = (lane >> 4) * 2;       // lanes 16..31 carry K+2,K+3

  v8f acc = {};
  const float* arow = A + (long)rcl * K + khalf;
  for (int k0 = 0; k0 < K; k0 += 4) {
    v2f a;
    a.x = arow[k0 + 0];                    // K = k0+khalf   (contiguous pair)
    a.y = arow[k0 + 1];                    // K = k0+khalf+1
    const float* bp = B + (long)(k0 + khalf) * Ncol + col;
    v2f b;
    b.x = bp[0];                           // same K rows of B, this column
    b.y = bp[Ncol];
    acc = __builtin_amdgcn_wmma_f32_16x16x4_f32(
        /*neg_a=*/false, a, /*neg_b=*/false, b,
        /*c_mod=*/(short)0, acc, /*reuse_a=*/false, /*reuse_b=*/false);
  }

  const int rbase = tm * 16 + (lane >> 4) * 8;  // D layout: VGPR v -> row rbase+v
#pragma unroll
  for (int v = 0; v < 8; ++v) {
    int r = rbase + v;
    if (r < M) C[(long)r * Ncol + col] = acc[v];
  }
}

// ---------------- attention logits: al = sum_c h[n,h,c] * a[h,c] -----------
__global__ void attn_logits(const float* __restrict__ h,
                            const float* __restrict__ as_,
                            const float* __restrict__ ad_,
                            float* __restrict__ al_s,
                            float* __restrict__ al_d, int Nn) {
  long i = (long)blockIdx.x * blockDim.x + threadIdx.x;   // over N*H
  if (i >= (long)Nn * HEADS) return;
  int n = (int)(i >> 2), hh = (int)(i & 3);
  const float* hp = h + (long)n * HC + hh * CDIM;
  const float* sp = as_ + hh * CDIM;
  const float* dp = ad_ + hh * CDIM;
  float s = 0.f, d = 0.f;
#pragma unroll 8
  for (int c = 0; c < CDIM; ++c) {
    float v = hp[c];
    s += v * sp[c];
    d += v * dp[c];
  }
  al_s[i] = s;
  al_d[i] = d;
}

// ---------------- edge pass 1: e = leakyrelu(al_s[src]+al_d[dst]); max -----
__global__ void edge_pass1(const int* __restrict__ srcA, const int* __restrict__ dstA,
                           long E0, long Etot,
                           const float* __restrict__ al_s, const float* __restrict__ al_d,
                           float* __restrict__ eedge, unsigned* __restrict__ mkey) {
  long i = (long)blockIdx.x * blockDim.x + threadIdx.x;   // over Etot*H
  if (i >= Etot * HEADS) return;
  long ee = i >> 2; int hh = (int)(i & 3);
  int s, d;
  if (ee < E0) { s = srcA[ee]; d = dstA[ee]; }
  else         { s = d = (int)(ee - E0); }                 // self loop
  float e = al_s[(long)s * HEADS + hh] + al_d[(long)d * HEADS + hh];
  e = e > 0.f ? e : 0.2f * e;                              // leaky_relu(0.2)
  eedge[i] = e;
  atomicMax(&mkey[(long)d * HEADS + hh], f2key(e));
}

// ---------------- edge pass 2: ex = exp(e - m[dst]); segment sum -----------
__global__ void edge_pass2(const int* __restrict__ dstA,
                           long E0, long Etot,
                           float* __restrict__ eedge,
                           const unsigned* __restrict__ mkey,
                           float* __restrict__ ssum) {
  long i = (long)blockIdx.x * blockDim.x + threadIdx.x;
  if (i >= Etot * HEADS) return;
  long ee = i >> 2; int hh = (int)(i & 3);
  int d = (ee < E0) ? dstA[ee] : (int)(ee - E0);
  float m = key2f(mkey[(long)d * HEADS + hh]);             // finite (self loop)
  float ex = expf(eedge[i] - m);
  eedge[i] = ex;
  atomicAdd(&ssum[(long)d * HEADS + hh], ex);
}

// ---------------- edge pass 3: out[dst] += h[src] * alpha ------------------
// One 256-thread block per edge; thread t covers head t/64, channel t%64.
__global__ void edge_message(const int* __restrict__ srcA, const int* __restrict__ dstA,
                             long E0,
                             const float* __restrict__ hin,
                             const float* __restrict__ eedge,
                             const float* __restrict__ ssum,
                             float* __restrict__ hagg) {
  long ee = blockIdx.x;
  int t = threadIdx.x;
  int hh = t >> 6;
  int s, d;
  if (ee < E0) { s = srcA[ee]; d = dstA[ee]; }
  else         { s = d = (int)(ee - E0); }
  float alpha = eedge[ee * HEADS + hh] /
                (ssum[(long)d * HEADS + hh] + 1e-16f);
  atomicAdd(&hagg[(long)d * HC + t], hin[(long)s * HC + t] * alpha);
}

// ---------------- bias + ELU (in place) ------------------------------------
__global__ void bias_elu(float* __restrict__ h, const float* __restrict__ b, long total) {
  long i = (long)blockIdx.x * blockDim.x + threadIdx.x;
  if (i >= total) return;
  float v = h[i] + b[i & (HC - 1)];
  h[i] = v > 0.f ? v : (expf(v) - 1.f);
}

// ---------------- global mean pool: atomic sums + counts -------------------
__global__ void pool_sum(const float* __restrict__ h, const int* __restrict__ batch,
                         float* __restrict__ pooled, float* __restrict__ cnt, int Nn) {
  int n = blockIdx.x;
  if (n >= Nn) return;
  int g = batch[n];
  atomicAdd(&pooled[(long)g * HC + threadIdx.x], h[(long)n * HC + threadIdx.x]);
  if (threadIdx.x == 0) atomicAdd(&cnt[g], 1.0f);
}

// ---------------- projection MLP -------------------------------------------
__global__ void mlp1(const float* __restrict__ pooled, const float* __restrict__ cnt,
                     const float* __restrict__ Wp1, const float* __restrict__ bp1,
                     float* __restrict__ z) {
  long i = (long)blockIdx.x * blockDim.x + threadIdx.x;    // G*512
  if (i >= (long)NGRAPH * 512) return;
  int g = (int)(i >> 9), j = (int)(i & 511);
  float inv = 1.0f / fmaxf(cnt[g], 1.0f);
  float acc = bp1[j];
  for (int k = 0; k < HC; ++k)
    acc += pooled[(long)g * HC + k] * inv * Wp1[(long)k * 512 + j];
  z[i] = fmaxf(acc, 0.0f);
}

__global__ void mlp2(const float* __restrict__ z,
                     const float* __restrict__ Wp2, const float* __restrict__ bp2,
                     float* __restrict__ out) {
  long i = (long)blockIdx.x * blockDim.x + threadIdx.x;    // G*256
  if (i >= (long)NGRAPH * HC) return;
  int g = (int)(i >> 8), j = (int)(i & 255);
  float acc = bp2[j];
  for (int k = 0; k < 512; ++k)
    acc += z[(long)g * 512 + k] * Wp2[(long)k * HC + j];
  out[i] = acc;
}

// ===========================================================================
extern "C" void kernel_launch(void* const* d_in, const int* in_sizes, int n_in,
                              void* d_out, int out_size, void* d_ws, size_t ws_size,
                              hipStream_t stream) {
  const float* x    = (const float*)d_in[0];
  const int*   ei   = (const int*)  d_in[1];
  const int*   batch= (const int*)  d_in[2];
  const float* W1   = (const float*)d_in[3];
  const float* a1s  = (const float*)d_in[4];
  const float* a1d  = (const float*)d_in[5];
  const float* b1   = (const float*)d_in[6];
  const float* W2   = (const float*)d_in[7];
  const float* a2s  = (const float*)d_in[8];
  const float* a2d  = (const float*)d_in[9];
  const float* b2   = (const float*)d_in[10];
  const float* W3   = (const float*)d_in[11];
  const float* a3s  = (const float*)d_in[12];
  const float* a3d  = (const float*)d_in[13];
  const float* b3   = (const float*)d_in[14];
  const float* Wp1  = (const float*)d_in[15];
  const float* bp1  = (const float*)d_in[16];
  const float* Wp2  = (const float*)d_in[17];
  const float* bp2  = (const float*)d_in[18];

  const int  N    = in_sizes[0] / 2;
  const long E0   = in_sizes[1] / 2;
  const long Etot = E0 + N;
  const int* srcA = ei;          // edge_index row 0
  const int* dstA = ei + E0;     // edge_index row 1

  // ---- workspace carve-out ----
  char*  base = (char*)d_ws;
  size_t off  = 0;
  auto carve = [&](size_t bytes) -> char* {
    char* p = base + off;
    off = (off + bytes + 255) & ~(size_t)255;
    return p;
  };
  float*    buf0   = (float*)   carve((size_t)N * HC * 4);      // transform out
  float*    buf1   = (float*)   carve((size_t)N * HC * 4);      // aggregate out
  float*    al_s   = (float*)   carve((size_t)N * HEADS * 4);
  float*    al_d   = (float*)   carve((size_t)N * HEADS * 4);
  unsigned* mkey   = (unsigned*)carve((size_t)N * HEADS * 4);
  float*    ssum   = (float*)   carve((size_t)N * HEADS * 4);
  float*    eedge  = (float*)   carve((size_t)Etot * HEADS * 4);
  float*    pooled = (float*)   carve((size_t)NGRAPH * HC * 4);
  float*    cnt    = (float*)   carve((size_t)NGRAPH * 4);
  float*    zhid   = (float*)   carve((size_t)NGRAPH * 512 * 4);
  (void)ws_size;

  const int T = 256;
  const long nodeElems = (long)N * HC;
  const long edgeElems = Etot * HEADS;
  const int  nodeBlk   = (int)((nodeElems + T - 1) / T);
  const int  nhBlk     = (int)(((long)N * HEADS + T - 1) / T);
  const int  edgeBlk   = (int)((edgeElems + T - 1) / T);
  const int  gemmBlk   = (((N + 15) / 16) * (HC / 16) + 7) / 8;

  const float* Ws[3]  = {W1, W2, W3};
  const float* As[3]  = {a1s, a2s, a3s};
  const float* Ad[3]  = {a1d, a2d, a3d};
  const float* Bs[3]  = {b1, b2, b3};

  for (int L = 0; L < 3; ++L) {
    // transform: buf0 = h_in @ W
    if (L == 0) {
      layer1_transform<<<nodeBlk, T, 0, stream>>>(x, W1, buf0, N);
    } else {
      gemm_f32_wmma<<<gemmBlk, T, 0, stream>>>(buf1, Ws[L], buf0, N, HC, HC);
    }
    // reset softmax state + aggregation target
    fill_u32<<<nhBlk,  T, 0, stream>>>(mkey, 0u, (long)N * HEADS);
    fill_u32<<<nhBlk,  T, 0, stream>>>((unsigned*)ssum, 0u, (long)N * HEADS);
    fill_u32<<<nodeBlk,T, 0, stream>>>((unsigned*)buf1, 0u, nodeElems);
    // attention
    attn_logits<<<nhBlk, T, 0, stream>>>(buf0, As[L], Ad[L], al_s, al_d, N);
    edge_pass1<<<edgeBlk, T, 0, stream>>>(srcA, dstA, E0, Etot, al_s, al_d, eedge, mkey);
    edge_pass2<<<edgeBlk, T, 0, stream>>>(dstA, E0, Etot, eedge, mkey, ssum);
    edge_message<<<(int)Etot, T, 0, stream>>>(srcA, dstA, E0, buf0, eedge, ssum, buf1);
    // bias + ELU in place; buf1 becomes next layer's input
    bias_elu<<<nodeBlk, T, 0, stream>>>(buf1, Bs[L], nodeElems);
  }

  // global mean pool
  fill_u32<<<(NGRAPH * HC + T - 1) / T, T, 0, stream>>>((unsigned*)pooled, 0u, (long)NGRAPH * HC);
  fill_u32<<<1, T, 0, stream>>>((unsigned*)cnt, 0u, NGRAPH);
  pool_sum<<<N, T, 0, stream>>>(buf1, batch, pooled, cnt, N);

  // projection MLP
  mlp1<<<(NGRAPH * 512 + T - 1) / T, T, 0, stream>>>(pooled, cnt, Wp1, bp1, zhid);
  mlp2<<<(NGRAPH * HC + T - 1) / T, T, 0, stream>>>(zhid, Wp2, bp2, (float*)d_out);
}